// InverseConsistencyLoss_38019050504639
// MI455X (gfx1250) — compile-verified
//
#include <hip/hip_runtime.h>

// InverseConsistencyLoss on (B=2, 3, 128,128,128) fp32 deformation fields.
// Stage 1: block-uniform (direction, batch) decode -> scalar base addresses;
//          grid-stride trilinear warp + squared-error accumulation; per-wave
//          reduction with V_WMMA_F32_16X16X4_F32 (B = ones -> row sums of the
//          16x4 lane-partial matrix) + one shfl_xor(16).
// Stage 2: single-block deterministic tree reduction + mean scaling.

#define DHW   (128 * 128 * 128)   // 1<<21
#define HW    (128 * 128)         // 1<<14
#define NTHREADS 256
#define BLOCKS_PER_COMBO 1024     // 4 combos (dir x batch) -> 4096 blocks total
#define NBLOCKS (4 * BLOCKS_PER_COMBO)
#define ITERS 8                   // 1024*256*8 == 2^21 voxels per combo, exact

typedef __attribute__((ext_vector_type(2))) float v2f;
typedef __attribute__((ext_vector_type(8))) float v8f;

__device__ __forceinline__ float lerp1(float a, float b, float w) {
    // a*(1-w) + b*w, matching the reference formulation
    return fmaf(b, w, (1.0f - w) * a);
}

__global__ void __launch_bounds__(NTHREADS)
icl_partial_kernel(const float* __restrict__ fwd,
                   const float* __restrict__ bwd,
                   float* __restrict__ ws)
{
    const int tid = threadIdx.x;

    // Block-uniform decode: combo = (dir, b); all base pointers end up in SGPRs.
    const int combo = blockIdx.x >> 10;           // 0..3
    const int blk   = blockIdx.x & (BLOCKS_PER_COMBO - 1);
    const int dir   = combo >> 1;                 // 0: apply=fwd,warp=bwd
    const int b     = combo & 1;

    const float* __restrict__ apply = dir ? bwd : fwd;
    const float* __restrict__ warp  = dir ? fwd : bwd;
    const float* __restrict__ ab = apply + ((size_t)(3 * b) << 21);
    const float* __restrict__ wb = warp  + ((size_t)(3 * b) << 21);

    float acc = 0.0f;

    int v = blk * NTHREADS + tid;                 // voxel index within volume

    #pragma unroll 1
    for (int k = 0; k < ITERS; ++k, v += BLOCKS_PER_COMBO * NTHREADS) {
        const int z = v >> 14;
        const int y = (v >> 7) & 127;
        const int x = v & 127;

        const float a0 = ab[v];
        const float a1 = ab[v + DHW];
        const float a2 = ab[v + 2 * DHW];

        // Reference semantics: grid channel 0 -> W axis, 1 -> H, 2 -> D.
        // All dims are 128 with align_corners=True so unnormalization cancels.
        const float fx = fminf(fmaxf((float)z + a0, 0.0f), 127.0f); // W coord
        const float fy = fminf(fmaxf((float)y + a1, 0.0f), 127.0f); // H coord
        const float fz = fminf(fmaxf((float)x + a2, 0.0f), 127.0f); // D coord

        const float x0f = floorf(fx), y0f = floorf(fy), z0f = floorf(fz);
        const float wx = fx - x0f, wy = fy - y0f, wz = fz - z0f;
        const int x0 = (int)x0f, y0 = (int)y0f, z0 = (int)z0f;
        const int x1 = min(x0 + 1, 127);
        const int y1 = min(y0 + 1, 127);
        const int z1 = min(z0 + 1, 127);

        const int zy00 = z0 * HW + y0 * 128;
        const int zy01 = z0 * HW + y1 * 128;
        const int zy10 = z1 * HW + y0 * 128;
        const int zy11 = z1 * HW + y1 * 128;

        const float av[3] = {a0, a1, a2};
        #pragma unroll
        for (int c = 0; c < 3; ++c) {
            const float* __restrict__ p = wb + ((size_t)c << 21);
            const float c000 = p[zy00 + x0], c001 = p[zy00 + x1];
            const float c010 = p[zy01 + x0], c011 = p[zy01 + x1];
            const float c100 = p[zy10 + x0], c101 = p[zy10 + x1];
            const float c110 = p[zy11 + x0], c111 = p[zy11 + x1];
            const float c00 = lerp1(c000, c001, wx);
            const float c01 = lerp1(c010, c011, wx);
            const float c10 = lerp1(c100, c101, wx);
            const float c11 = lerp1(c110, c111, wx);
            const float c0  = lerp1(c00, c01, wy);
            const float c1  = lerp1(c10, c11, wy);
            const float val = lerp1(c0, c1, wz);
            const float e   = av[c] + val;
            acc = fmaf(e, e, acc);
        }
    }

    // ---- wave-level reduction via WMMA (f32 in/out, precision-safe) ----
    // A (16x4): lane L, VGPR0 -> A[L%16, L<16 ? 0 : 2] = acc, VGPR1 slots = 0.
    // B (4x16) = all ones -> D[m,n] = sum_k A[m,k] = acc_m + acc_{16+m}.
    v2f a;    a.x = acc;  a.y = 0.0f;
    v2f ones; ones.x = 1.0f; ones.y = 1.0f;
    v8f cacc = {};
    cacc = __builtin_amdgcn_wmma_f32_16x16x4_f32(
        /*neg_a=*/false, a, /*neg_b=*/false, ones,
        /*c_mod=*/(short)0, cacc, /*reuse_a=*/false, /*reuse_b=*/false);

    // Lanes 0-15 hold D rows 0..7 (col = lane); lanes 16-31 hold rows 8..15.
    float h = cacc[0] + cacc[1] + cacc[2] + cacc[3]
            + cacc[4] + cacc[5] + cacc[6] + cacc[7];
    h += __shfl_xor(h, 16, 32);   // combine the two half-wave row groups

    __shared__ float wsum[NTHREADS / 32];
    const int lane = tid & 31;
    const int wave = tid >> 5;
    if (lane == 0) wsum[wave] = h;
    __syncthreads();

    if (tid == 0) {
        float s = 0.0f;
        #pragma unroll
        for (int w = 0; w < NTHREADS / 32; ++w) s += wsum[w];  // fixed order
        ws[blockIdx.x] = s;
    }
}

__global__ void __launch_bounds__(256)
icl_final_kernel(const float* __restrict__ ws, float* __restrict__ out)
{
    __shared__ float sm[256];
    const int tid = threadIdx.x;
    float s = 0.0f;
    #pragma unroll
    for (int i = tid; i < NBLOCKS; i += 256) s += ws[i];  // 16 adds, fixed order
    sm[tid] = s;
    __syncthreads();
    #pragma unroll
    for (int step = 128; step > 0; step >>= 1) {
        if (tid < step) sm[tid] += sm[tid + step];
        __syncthreads();
    }
    if (tid == 0) {
        // Both direction-means share denominator N = 2*3*128^3 = 12,582,912.
        out[0] = sm[0] * (1.0f / 12582912.0f);
    }
}

extern "C" void kernel_launch(void* const* d_in, const int* in_sizes, int n_in,
                              void* d_out, int out_size, void* d_ws, size_t ws_size,
                              hipStream_t stream) {
    (void)in_sizes; (void)n_in; (void)out_size; (void)ws_size;
    const float* fwd = (const float*)d_in[0];
    const float* bwd = (const float*)d_in[1];
    float* ws  = (float*)d_ws;     // NBLOCKS floats = 16 KB of scratch
    float* out = (float*)d_out;

    icl_partial_kernel<<<NBLOCKS, NTHREADS, 0, stream>>>(fwd, bwd, ws);
    icl_final_kernel<<<1, 256, 0, stream>>>(ws, out);
}